// HELPv3_16509854286588
// MI455X (gfx1250) — compile-verified
//
#include <hip/hip_runtime.h>
#include <hip/hip_bf16.h>
#include <cmath>

// ---------------------------------------------------------------------------
// Problem constants (from the reference)
// ---------------------------------------------------------------------------
#define NH_ 50000
#define NO_ 50000
#define E_  500000
#define D_  256
#define L_  8
#define B_  512
#define C_  117

typedef float v2f __attribute__((ext_vector_type(2)));
typedef float v8f __attribute__((ext_vector_type(8)));

// ---------------------------------------------------------------------------
// FP32 WMMA GEMM:  out[N,256] = A[N,256] @ W[256,256]
// One 256-thread block = 8 waves. Wave w computes rows
//   rowTile = blockIdx.x*32 + (w>>2)*16   and cols colTile = (w&3)*64
// using V_WMMA_F32_16X16X4_F32 with 4 accumulators (A fragment reused).
// A fragment layout (16x4):  lane holds M = lane%16, VGPR j holds K = j + 2*(lane>>4)
// B fragment layout (4x16):  lane holds N = lane%16, VGPR j holds K = j + 2*(lane>>4)
// D layout (16x16):          lane holds N = lane%16, VGPR v holds M = v + 8*(lane>>4)
// ---------------------------------------------------------------------------
__global__ __launch_bounds__(256)
void gemm256_wmma(const float* __restrict__ A, const float* __restrict__ W,
                  float* __restrict__ out, int N)
{
    const int tid  = threadIdx.x;
    const int wave = tid >> 5;
    const int lane = tid & 31;
    const int l16  = lane & 15;
    const int lhi  = lane >> 4;                 // 0 or 1
    const int rowTile = blockIdx.x * 32 + (wave >> 2) * 16;
    const int colTile = (wave & 3) * 64;

    v8f acc0 = {}; v8f acc1 = {}; v8f acc2 = {}; v8f acc3 = {};

    int arow = rowTile + l16;
    if (arow >= N) arow = N - 1;                // clamp: safe load, store is guarded
    const float* __restrict__ aptr = A + (size_t)arow * D_;

    for (int k0 = 0; k0 < D_; k0 += 4) {
        const int ka = k0 + 2 * lhi;            // K for VGPR j: ka + j
        v2f a;
        a.x = aptr[ka];
        a.y = aptr[ka + 1];

        const float* __restrict__ w0 = W + (size_t)ka * D_ + colTile + l16;
        const float* __restrict__ w1 = w0 + D_;
        v2f b0, b1, b2, b3;
        b0.x = w0[0];   b0.y = w1[0];
        b1.x = w0[16];  b1.y = w1[16];
        b2.x = w0[32];  b2.y = w1[32];
        b3.x = w0[48];  b3.y = w1[48];

        acc0 = __builtin_amdgcn_wmma_f32_16x16x4_f32(false, a, false, b0, (short)0, acc0, false, false);
        acc1 = __builtin_amdgcn_wmma_f32_16x16x4_f32(false, a, false, b1, (short)0, acc1, false, false);
        acc2 = __builtin_amdgcn_wmma_f32_16x16x4_f32(false, a, false, b2, (short)0, acc2, false, false);
        acc3 = __builtin_amdgcn_wmma_f32_16x16x4_f32(false, a, false, b3, (short)0, acc3, false, false);
    }

#pragma unroll
    for (int v = 0; v < 8; ++v) {
        const int m = rowTile + v + 8 * lhi;
        if (m < N) {
            float* __restrict__ o = out + (size_t)m * D_ + colTile + l16;
            o[0]  = acc0[v];
            o[16] = acc1[v];
            o[32] = acc2[v];
            o[48] = acc3[v];
        }
    }
}

// ---------------------------------------------------------------------------
// Helpers
// ---------------------------------------------------------------------------
__device__ inline float waveReduceSum(float v) {
#pragma unroll
    for (int off = 16; off > 0; off >>= 1) v += __shfl_xor(v, off, 32);
    return v;
}

// float atomic max via ordered-integer trick (buffer always holds float bits)
__device__ inline void atomicMaxFloat(float* addr, float val) {
    if (val >= 0.0f) atomicMax((int*)addr, __float_as_int(val));
    else             atomicMin((unsigned int*)addr, __float_as_uint(val));
}

__global__ void fillf(float* __restrict__ p, float v, int n) {
    int i = blockIdx.x * 256 + threadIdx.x;
    if (i < n) p[i] = v;
}

// out[n] = dot(X[n,:], vec)   (wave per row)
__global__ void rowdot256(const float* __restrict__ X, const float* __restrict__ vec,
                          float* __restrict__ out, int N)
{
    const int wave = threadIdx.x >> 5, lane = threadIdx.x & 31;
    const int row = blockIdx.x * 8 + wave;
    if (row >= N) return;
    const float* __restrict__ x = X + (size_t)row * D_;
    float s = 0.f;
#pragma unroll
    for (int i = 0; i < 8; ++i) s += x[lane + 32 * i] * vec[lane + 32 * i];
    s = waveReduceSum(s);
    if (lane == 0) out[row] = s;
}

// out[k] = dot(M[k,:], v)   (one 256-thread block)
__global__ void matvec256(const float* __restrict__ M, const float* __restrict__ v,
                          float* __restrict__ out)
{
    const int k = threadIdx.x;
    float s = 0.f;
    for (int j = 0; j < D_; ++j) s += M[(size_t)k * D_ + j] * v[j];
    out[k] = s;
}

// we[r] = dot(Wedge[r,:], aedge) for r=0,1   (one 256-thread block)
__global__ void edgeProj(const float* __restrict__ Wedge, const float* __restrict__ aedge,
                         float* __restrict__ we)
{
    __shared__ float s0[256], s1[256];
    const int t = threadIdx.x;
    s0[t] = Wedge[t] * aedge[t];
    s1[t] = Wedge[D_ + t] * aedge[t];
    __syncthreads();
    for (int off = 128; off > 0; off >>= 1) {
        if (t < off) { s0[t] += s0[t + off]; s1[t] += s1[t + off]; }
        __syncthreads();
    }
    if (t == 0) { we[0] = s0[0]; we[1] = s1[0]; }
}

// pass 1: logit = leaky_relu(a_s[src] + a_d[dst] + a_e, 0.2); segment max over dst
__global__ void edgeLogitMax(const int* __restrict__ src, const int* __restrict__ dst,
                             const float* __restrict__ ea, const float* __restrict__ we,
                             const float* __restrict__ a_s, const float* __restrict__ a_d,
                             float* __restrict__ logit, float* __restrict__ maxb, int E)
{
    const int e = blockIdx.x * 256 + threadIdx.x;
    if (e >= E) return;
    const int s = src[e], d = dst[e];
    const float ae = ea[2 * e] * we[0] + ea[2 * e + 1] * we[1];
    float l = a_s[s] + a_d[d] + ae;
    l = (l > 0.f) ? l : 0.2f * l;
    logit[e] = l;
    atomicMaxFloat(&maxb[d], l);
}

// pass 2: ex = exp(logit - max[dst]); segment sum over dst
__global__ void edgeExpSum(const int* __restrict__ dst, const float* __restrict__ logit,
                           const float* __restrict__ maxb, float* __restrict__ ex,
                           float* __restrict__ denom, int E)
{
    const int e = blockIdx.x * 256 + threadIdx.x;
    if (e >= E) return;
    const int d = dst[e];
    const float v = expf(logit[e] - maxb[d]);
    ex[e] = v;
    atomicAdd(&denom[d], v);
}

// pass 3: agg[dst,:] += h[src,:] * (ex/denom[dst])   (wave per edge)
__global__ void edgeScatter(const int* __restrict__ src, const int* __restrict__ dst,
                            const float* __restrict__ H, const float* __restrict__ ex,
                            const float* __restrict__ denom, float* __restrict__ agg, int E)
{
    const int wave = threadIdx.x >> 5, lane = threadIdx.x & 31;
    const int e = blockIdx.x * 8 + wave;
    if (e >= E) return;
    const int s = src[e], d = dst[e];
    const float alpha = ex[e] / denom[d];
    const float* __restrict__ h = H + (size_t)s * D_;
    float* __restrict__ o = agg + (size_t)d * D_;
#pragma unroll
    for (int i = 0; i < 8; ++i) {
        const int f = lane + 32 * i;
        atomicAdd(&o[f], h[f] * alpha);
    }
}

// x = relu(agg + bias) + (residual ? x : 0)     (in place)
__global__ void finalizeNode(const float* __restrict__ agg, const float* __restrict__ bias,
                             float* __restrict__ x, int N, int residual)
{
    const int i = blockIdx.x * 256 + threadIdx.x;
    if (i >= N * D_) return;
    const int f = i & (D_ - 1);
    float v = agg[i] + bias[f];
    v = (v > 0.f) ? v : 0.f;
    x[i] = v + (residual ? x[i] : 0.f);
}

// segment sum of node features + counts   (wave per node)
__global__ void nodePool(const float* __restrict__ X, const int* __restrict__ batch,
                         float* __restrict__ sum, float* __restrict__ cnt, int N)
{
    const int wave = threadIdx.x >> 5, lane = threadIdx.x & 31;
    const int n = blockIdx.x * 8 + wave;
    if (n >= N) return;
    const int b = batch[n];
    const float* __restrict__ x = X + (size_t)n * D_;
    float* __restrict__ s = sum + (size_t)b * D_;
#pragma unroll
    for (int i = 0; i < 8; ++i) atomicAdd(&s[lane + 32 * i], x[lane + 32 * i]);
    if (lane == 0) atomicAdd(&cnt[b], 1.0f);
}

// edge MLP (relu(ea @ W_emlp + b)) pooled by hbatch[src]   (wave per edge, 32 feats)
__global__ void edgePool(const float* __restrict__ ea, const int* __restrict__ src,
                         const int* __restrict__ hbatch, const float* __restrict__ Wm,
                         const float* __restrict__ bm, float* __restrict__ esum,
                         float* __restrict__ ecnt, int E)
{
    const int wave = threadIdx.x >> 5, lane = threadIdx.x & 31;   // lane == feature
    const int e = blockIdx.x * 8 + wave;
    if (e >= E) return;
    const int b = hbatch[src[e]];
    float v = ea[2 * e] * Wm[lane] + ea[2 * e + 1] * Wm[32 + lane] + bm[lane];
    v = (v > 0.f) ? v : 0.f;
    atomicAdd(&esum[b * 32 + lane], v);
    if (lane == 0) atomicAdd(&ecnt[b], 1.0f);
}

// emb[b, 0:256]=hmean, [256:512]=omean, [512:544]=emean
__global__ void buildEmb(const float* __restrict__ hsum, const float* __restrict__ hcnt,
                         const float* __restrict__ osum, const float* __restrict__ ocnt,
                         const float* __restrict__ esum, const float* __restrict__ ecnt,
                         float* __restrict__ emb)
{
    const int i = blockIdx.x * 256 + threadIdx.x;
    if (i >= B_ * 544) return;
    const int b = i / 544, j = i % 544;
    float v;
    if (j < 256)      v = hsum[b * 256 + j]        / fmaxf(hcnt[b], 1.f);
    else if (j < 512) v = osum[b * 256 + (j - 256)] / fmaxf(ocnt[b], 1.f);
    else              v = esum[b * 32  + (j - 512)] / fmaxf(ecnt[b], 1.f);
    emb[i] = v;
}

// one block per graph: two softmax heads over 117 classes
__global__ __launch_bounds__(128)
void classifyHeads(const float* __restrict__ emb,
                   const float* __restrict__ W1, const float* __restrict__ b1,
                   const float* __restrict__ W2, const float* __restrict__ b2,
                   float* __restrict__ out)
{
    __shared__ float se[544];
    __shared__ float red[128];
    const int b = blockIdx.x, t = threadIdx.x;
    for (int j = t; j < 544; j += 128) se[j] = emb[(size_t)b * 544 + j];
    __syncthreads();
    for (int head = 0; head < 2; ++head) {
        const float* __restrict__ W  = head ? W2 : W1;
        const float* __restrict__ bb = head ? b2 : b1;
        float logit = -3.0e38f;
        if (t < C_) {
            float s = bb[t];
            for (int j = 0; j < 544; ++j) s += se[j] * W[j * C_ + t];
            logit = s;
        }
        red[t] = logit; __syncthreads();
        for (int off = 64; off > 0; off >>= 1) {
            if (t < off) red[t] = fmaxf(red[t], red[t + off]);
            __syncthreads();
        }
        const float m = red[0]; __syncthreads();
        const float ev = (t < C_) ? expf(logit - m) : 0.f;
        red[t] = ev; __syncthreads();
        for (int off = 64; off > 0; off >>= 1) {
            if (t < off) red[t] += red[t + off];
            __syncthreads();
        }
        const float ssum = red[0]; __syncthreads();
        if (t < C_) out[(size_t)b * (2 * C_) + head * C_ + t] = ev / ssum;
    }
}

// ---------------------------------------------------------------------------
// Host orchestration
// ---------------------------------------------------------------------------
static inline int cdiv_i(int a, int b) { return (a + b - 1) / b; }

extern "C" void kernel_launch(void* const* d_in, const int* in_sizes, int n_in,
                              void* d_out, int out_size, void* d_ws, size_t ws_size,
                              hipStream_t stream)
{
    (void)in_sizes; (void)n_in; (void)out_size; (void)ws_size;
    const float* x_human  = (const float*)d_in[0];
    const float* x_object = (const float*)d_in[1];
    const int*   ei_ho    = (const int*)d_in[2];
    const int*   ei_oh    = (const int*)d_in[3];
    const float* ea_ho    = (const float*)d_in[4];
    const float* ea_oh    = (const float*)d_in[5];
    const int*   hbatch   = (const int*)d_in[6];
    const int*   obatch   = (const int*)d_in[7];
    const float* Wsrc_ho  = (const float*)d_in[8];
    const float* Wdst_ho  = (const float*)d_in[9];
    const float* asrc_ho  = (const float*)d_in[10];
    const float* adst_ho  = (const float*)d_in[11];
    const float* Wedge_ho = (const float*)d_in[12];
    const float* aedge_ho = (const float*)d_in[13];
    const float* bias_ho  = (const float*)d_in[14];
    const float* Wsrc_oh  = (const float*)d_in[15];
    const float* Wdst_oh  = (const float*)d_in[16];
    const float* asrc_oh  = (const float*)d_in[17];
    const float* adst_oh  = (const float*)d_in[18];
    const float* Wedge_oh = (const float*)d_in[19];
    const float* aedge_oh = (const float*)d_in[20];
    const float* bias_oh  = (const float*)d_in[21];
    const float* W_emlp   = (const float*)d_in[22];
    const float* b_emlp   = (const float*)d_in[23];
    const float* W_p1     = (const float*)d_in[24];
    const float* b_p1     = (const float*)d_in[25];
    const float* W_p2     = (const float*)d_in[26];
    const float* b_p2     = (const float*)d_in[27];

    // ---- workspace carve-up (floats) ----
    float* w = (float*)d_ws;
    size_t off = 0;
    auto carve = [&](size_t n) { float* p = w + off; off += n; return p; };
    float* xh    = carve((size_t)NH_ * D_);
    float* xo    = carve((size_t)NO_ * D_);
    float* hbuf  = carve((size_t)NH_ * D_);
    float* aggO  = carve((size_t)NO_ * D_);
    float* aggH  = carve((size_t)NH_ * D_);
    float* a_s   = carve(NH_);
    float* a_d   = carve(NH_);
    float* logit = carve(E_);
    float* ex    = carve(E_);
    float* maxb  = carve(NH_);
    float* denom = carve(NH_);
    float* wvec  = carve(D_);
    float* wePrj = carve(2);
    float* hsum  = carve((size_t)B_ * D_);
    float* osum  = carve((size_t)B_ * D_);
    float* esum  = carve((size_t)B_ * 32);
    float* hcnt  = carve(B_);
    float* ocnt  = carve(B_);
    float* ecnt  = carve(B_);
    float* emb   = carve((size_t)B_ * 544);

    const int* src_ho = ei_ho;      const int* dst_ho = ei_ho + E_;
    const int* src_oh = ei_oh;      const int* dst_oh = ei_oh + E_;

    const float NEG_INF = -__builtin_huge_valf();

    hipMemcpyAsync(xh, x_human,  (size_t)NH_ * D_ * sizeof(float), hipMemcpyDeviceToDevice, stream);
    hipMemcpyAsync(xo, x_object, (size_t)NO_ * D_ * sizeof(float), hipMemcpyDeviceToDevice, stream);

    for (int l = 0; l < L_; ++l) {
        const int res = (l > 0) ? 1 : 0;

        // ---------- relation ho: src=xh(human) -> dst=xo(object), output aggO ----------
        gemm256_wmma<<<cdiv_i(NH_, 32), 256, 0, stream>>>(xh, Wsrc_ho + (size_t)l * D_ * D_, hbuf, NH_);
        rowdot256<<<cdiv_i(NH_, 8), 256, 0, stream>>>(hbuf, asrc_ho + l * D_, a_s, NH_);
        matvec256<<<1, 256, 0, stream>>>(Wdst_ho + (size_t)l * D_ * D_, adst_ho + l * D_, wvec);
        rowdot256<<<cdiv_i(NO_, 8), 256, 0, stream>>>(xo, wvec, a_d, NO_);
        edgeProj<<<1, 256, 0, stream>>>(Wedge_ho + (size_t)l * 2 * D_, aedge_ho + l * D_, wePrj);
        fillf<<<cdiv_i(NO_, 256), 256, 0, stream>>>(maxb, NEG_INF, NO_);
        fillf<<<cdiv_i(NO_, 256), 256, 0, stream>>>(denom, 0.f, NO_);
        fillf<<<cdiv_i(NO_ * D_, 256), 256, 0, stream>>>(aggO, 0.f, NO_ * D_);
        edgeLogitMax<<<cdiv_i(E_, 256), 256, 0, stream>>>(src_ho, dst_ho, ea_ho, wePrj, a_s, a_d, logit, maxb, E_);
        edgeExpSum<<<cdiv_i(E_, 256), 256, 0, stream>>>(dst_ho, logit, maxb, ex, denom, E_);
        edgeScatter<<<cdiv_i(E_, 8), 256, 0, stream>>>(src_ho, dst_ho, hbuf, ex, denom, aggO, E_);

        // ---------- relation oh: src=xo(object) -> dst=xh(human), output aggH ----------
        gemm256_wmma<<<cdiv_i(NO_, 32), 256, 0, stream>>>(xo, Wsrc_oh + (size_t)l * D_ * D_, hbuf, NO_);
        rowdot256<<<cdiv_i(NO_, 8), 256, 0, stream>>>(hbuf, asrc_oh + l * D_, a_s, NO_);
        matvec256<<<1, 256, 0, stream>>>(Wdst_oh + (size_t)l * D_ * D_, adst_oh + l * D_, wvec);
        rowdot256<<<cdiv_i(NH_, 8), 256, 0, stream>>>(xh, wvec, a_d, NH_);
        edgeProj<<<1, 256, 0, stream>>>(Wedge_oh + (size_t)l * 2 * D_, aedge_oh + l * D_, wePrj);
        fillf<<<cdiv_i(NH_, 256), 256, 0, stream>>>(maxb, NEG_INF, NH_);
        fillf<<<cdiv_i(NH_, 256), 256, 0, stream>>>(denom, 0.f, NH_);
        fillf<<<cdiv_i(NH_ * D_, 256), 256, 0, stream>>>(aggH, 0.f, NH_ * D_);
        edgeLogitMax<<<cdiv_i(E_, 256), 256, 0, stream>>>(src_oh, dst_oh, ea_oh, wePrj, a_s, a_d, logit, maxb, E_);
        edgeExpSum<<<cdiv_i(E_, 256), 256, 0, stream>>>(dst_oh, logit, maxb, ex, denom, E_);
        edgeScatter<<<cdiv_i(E_, 8), 256, 0, stream>>>(src_oh, dst_oh, hbuf, ex, denom, aggH, E_);

        // ---------- ReLU + bias + residual (in place) ----------
        finalizeNode<<<cdiv_i(NO_ * D_, 256), 256, 0, stream>>>(aggO, bias_ho + l * D_, xo, NO_, res);
        finalizeNode<<<cdiv_i(NH_ * D_, 256), 256, 0, stream>>>(aggH, bias_oh + l * D_, xh, NH_, res);
    }

    // ---------- pooling ----------
    fillf<<<cdiv_i(B_ * D_, 256), 256, 0, stream>>>(hsum, 0.f, B_ * D_);
    fillf<<<cdiv_i(B_ * D_, 256), 256, 0, stream>>>(osum, 0.f, B_ * D_);
    fillf<<<cdiv_i(B_ * 32, 256), 256, 0, stream>>>(esum, 0.f, B_ * 32);
    fillf<<<cdiv_i(B_, 256), 256, 0, stream>>>(hcnt, 0.f, B_);
    fillf<<<cdiv_i(B_, 256), 256, 0, stream>>>(ocnt, 0.f, B_);
    fillf<<<cdiv_i(B_, 256), 256, 0, stream>>>(ecnt, 0.f, B_);
    nodePool<<<cdiv_i(NH_, 8), 256, 0, stream>>>(xh, hbatch, hsum, hcnt, NH_);
    nodePool<<<cdiv_i(NO_, 8), 256, 0, stream>>>(xo, obatch, osum, ocnt, NO_);
    edgePool<<<cdiv_i(E_, 8), 256, 0, stream>>>(ea_ho, src_ho, hbatch, W_emlp, b_emlp, esum, ecnt, E_);
    buildEmb<<<cdiv_i(B_ * 544, 256), 256, 0, stream>>>(hsum, hcnt, osum, ocnt, esum, ecnt, emb);

    // ---------- classifier heads ----------
    classifyHeads<<<B_, 128, 0, stream>>>(emb, W_p1, b_p1, W_p2, b_p2, (float*)d_out);
}